// Freeness_72894184947911
// MI455X (gfx1250) — compile-verified
//
#include <hip/hip_runtime.h>
#include <stdint.h>

// MI455X / gfx1250, wave32. One 1024-thread workgroup per batch row.
// Dynamic LDS: 16384 x u64 sort buffer (128KB); lower 64KB overlaid as the
// original-order allocation buffer after the sorted reads are staged in regs.

typedef float v2f __attribute__((ext_vector_type(2)));
typedef float v8f __attribute__((ext_vector_type(8)));

#define MDIM   16384
#define BDIM   1024
#define CHUNK  16
#define NCHUNK 1024   // MDIM / CHUNK
#define NROW   64     // NCHUNK / 16

__global__ __launch_bounds__(1024)
void dnc_freeness_kernel(const float* __restrict__ ww,   // [B,4,M]
                         const float* __restrict__ fg,   // [B,8]
                         const float* __restrict__ rw,   // [B,8,M]
                         const float* __restrict__ pu,   // [B,M]
                         const float* __restrict__ wg,   // [B,4]
                         float* __restrict__ out_usage,  // [B,M]
                         float* __restrict__ out_alloc)  // [B,4,M]
{
  extern __shared__ unsigned long long sbuf[];   // 16384 u64 (128 KB)
  float* allocf = (float*)sbuf;                  // overlay (first 64 KB)

  __shared__ float Lmat[NCHUNK];   // log2 of per-chunk products (64x16 matrix)
  __shared__ float Wmat[NCHUNK];   // exclusive within-row prefix sums
  __shared__ float rowInc[NROW];
  __shared__ float rowOff[NROW];

  const int b  = blockIdx.x;
  const int t  = threadIdx.x;
  const int m0 = t * CHUNK;

  // ---------------- Phase 1: usage after write + free (streaming) ----------
  float u[CHUNK];
  {
    float fgv[8];
#pragma unroll
    for (int r = 0; r < 8; ++r) fgv[r] = fg[(size_t)b * 8 + r];

    const size_t mbase = (size_t)b * MDIM + m0;
#pragma unroll
    for (int q = 0; q < 4; ++q) {
      float puv[4], pw[4], ph[4], ou[4];
      *(float4*)puv = *(const float4*)(pu + mbase + 4 * q);
#pragma unroll
      for (int c = 0; c < 4; ++c) { pw[c] = 1.0f; ph[c] = 1.0f; }
#pragma unroll
      for (int w = 0; w < 4; ++w) {
        float wv[4];
        *(float4*)wv = *(const float4*)(ww + ((size_t)b * 4 + w) * MDIM + m0 + 4 * q);
#pragma unroll
        for (int c = 0; c < 4; ++c) pw[c] *= (1.0f - wv[c]);
      }
#pragma unroll
      for (int r = 0; r < 8; ++r) {
        float rv[4];
        *(float4*)rv = *(const float4*)(rw + ((size_t)b * 8 + r) * MDIM + m0 + 4 * q);
#pragma unroll
        for (int c = 0; c < 4; ++c) ph[c] *= (1.0f - fgv[r] * rv[c]);
      }
#pragma unroll
      for (int c = 0; c < 4; ++c) {
        float uu = puv[c] + (1.0f - puv[c]) * (1.0f - pw[c]);
        uu *= ph[c];
        u[4 * q + c] = uu;
        ou[c] = uu;
      }
      *(float4*)(out_usage + mbase + 4 * q) = *(float4*)ou;
    }
  }

  const float ome = 1.0f - 1e-6f;   // (1 - epsilon)

  // ---------------- Phase 2: 4 write heads, chained allocation -------------
  for (int head = 0; head < 4; ++head) {
    const float wgh = wg[(size_t)b * 4 + head];

    __syncthreads();   // prior-head LDS consumers done (asynccnt already drained)

    // build (nonusage_bits << 32 | index) pairs
#pragma unroll
    for (int j = 0; j < CHUNK; ++j) {
      float nonus = ome * (1.0f - u[j]);         // = 1 - (eps + (1-eps)*u)
      nonus = fmaxf(nonus, 0.0f);                // keep IEEE-bit order valid
      unsigned key = __float_as_uint(nonus);
      sbuf[m0 + j] = ((unsigned long long)key << 32) | (unsigned)(m0 + j);
    }

    // bitonic sort, descending (== nonusage descending, i.e. usage ascending)
    for (int len = 2; len <= MDIM; len <<= 1) {
      for (int stride = len >> 1; stride > 0; stride >>= 1) {
        __syncthreads();
#pragma unroll
        for (int e = 0; e < CHUNK; ++e) {
          int i = (e << 10) | t;
          int ixj = i ^ stride;
          if (ixj > i) {
            unsigned long long va = sbuf[i];
            unsigned long long vb = sbuf[ixj];
            bool desc = ((i & len) == 0);
            bool sw = desc ? (va < vb) : (va > vb);
            if (sw) { sbuf[i] = vb; sbuf[ixj] = va; }
          }
        }
      }
    }
    __syncthreads();

    // pass A over this thread's 16 sorted positions: chunk log2-product only
    // (nothing kept live across the WMMA scan except u[16])
    {
      float run = 1.0f;
#pragma unroll
      for (int j = 0; j < CHUNK; ++j) {
        unsigned long long pr = sbuf[m0 + j];
        float nonus = __uint_as_float((unsigned)(pr >> 32));
        run *= (1.0f - nonus);                   // sorted_usage
      }
      Lmat[t] = __log2f(run);                    // chunk log2-product
    }
    __syncthreads();

    // inter-chunk exclusive scan of 1024 log2-products:
    // view as 64x16 matrix; W = L x T (T strictly upper-triangular ones)
#if defined(__has_builtin) && __has_builtin(__builtin_amdgcn_wmma_f32_16x16x4_f32)
    if (t < 32) {
      const int lane = t;
      const int n    = lane & 15;   // A-row for this lane AND B/D-column
      const int half = lane >> 4;
#pragma unroll
      for (int tile = 0; tile < 4; ++tile) {
        v8f d = {0.f, 0.f, 0.f, 0.f, 0.f, 0.f, 0.f, 0.f};
#pragma unroll
        for (int kk = 0; kk < 4; ++kk) {
          const int k0 = 4 * kk + 2 * half;      // K index of .x component
          v2f a, bm;
          a.x  = Lmat[(16 * tile + n) * 16 + (k0 + 0)];
          a.y  = Lmat[(16 * tile + n) * 16 + (k0 + 1)];
          bm.x = ((k0 + 0) < n) ? 1.0f : 0.0f;   // strictly upper: k < col
          bm.y = ((k0 + 1) < n) ? 1.0f : 0.0f;
          d = __builtin_amdgcn_wmma_f32_16x16x4_f32(
                  false, a, false, bm, (short)0, d, false, false);
        }
#pragma unroll
        for (int r = 0; r < 8; ++r) {
          const int rowi = 16 * tile + r + 8 * half;
          Wmat[rowi * 16 + n] = d[r];
          if (n == 15) rowInc[rowi] = d[r] + Lmat[rowi * 16 + 15];
        }
      }
    }
#else
    if (t < NROW) {                               // scalar fallback
      float run = 0.0f;
#pragma unroll
      for (int j = 0; j < 16; ++j) { Wmat[t * 16 + j] = run; run += Lmat[t * 16 + j]; }
      rowInc[t] = run;
    }
#endif
    __syncthreads();

    if (t == 0) {                                 // tiny scan over 64 rows
      float acc = 0.0f;
      for (int i = 0; i < NROW; ++i) { rowOff[i] = acc; acc += rowInc[i]; }
    }
    __syncthreads();

    // pass B: re-read sorted pairs (sort buffer still intact), rebuild the
    // 16-element exclusive prefix, then barrier before the overlaying scatter.
    unsigned idxs[CHUNK];
    float    aval[CHUNK];
    {
      const float S    = rowOff[t >> 4] + Wmat[t];
      const float base = exp2f(S);                // exclusive product of prior chunks
      float run = base;
#pragma unroll
      for (int j = 0; j < CHUNK; ++j) {
        unsigned long long pr = sbuf[m0 + j];
        idxs[j] = (unsigned)pr;
        float nonus = __uint_as_float((unsigned)(pr >> 32));
        aval[j] = nonus * run;                    // sorted_allocation
        run *= (1.0f - nonus);
      }
    }
    __syncthreads();   // all re-reads done before scatter overwrites sbuf

    // scatter to original order (overlays lower half of the sort buffer)
#pragma unroll
    for (int j = 0; j < CHUNK; ++j) {
      allocf[idxs[j]] = aval[j];
    }
    __syncthreads();

    // emit alloc row via async LDS->global DMA (ASYNCcnt-tracked), then
    // update usage registers for the next head.
    {
      const float* gout = out_alloc + ((size_t)b * 4 + head) * MDIM + m0;
      const unsigned lds0 = (unsigned)(uintptr_t)(&allocf[m0]);
#pragma unroll
      for (int q = 0; q < 4; ++q) {
        unsigned long long ga = (unsigned long long)(uintptr_t)(gout + 4 * q);
        unsigned la = lds0 + 16u * q;
        asm volatile("global_store_async_from_lds_b128 %0, %1, off"
                     :: "v"(ga), "v"(la) : "memory");
      }
#pragma unroll
      for (int j = 0; j < CHUNK; ++j) {
        float a = allocf[m0 + j];
        u[j] = u[j] + (1.0f - u[j]) * wgh * a;
      }
      // drain this wave's async stores before anyone rewrites allocf/sbuf
      asm volatile("s_wait_asynccnt 0" ::: "memory");
    }
  }
}

extern "C" void kernel_launch(void* const* d_in, const int* in_sizes, int n_in,
                              void* d_out, int out_size, void* d_ws, size_t ws_size,
                              hipStream_t stream) {
  const float* ww = (const float*)d_in[0];  // write_weights [B,4,M]
  const float* fg = (const float*)d_in[1];  // free_gate     [B,8]
  const float* rw = (const float*)d_in[2];  // read_weights  [B,8,M]
  const float* pu = (const float*)d_in[3];  // prev_usage    [B,M]
  const float* wg = (const float*)d_in[4];  // write_gates   [B,4]
  // d_in[5] = num_writes (== 4, fixed by the shapes)

  const int B = in_sizes[3] / MDIM;
  float* out_usage = (float*)d_out;
  float* out_alloc = out_usage + (size_t)B * MDIM;

  const size_t shmem = (size_t)MDIM * sizeof(unsigned long long);  // 128 KB
  dnc_freeness_kernel<<<dim3(B), dim3(BDIM), shmem, stream>>>(
      ww, fg, rw, pu, wg, out_usage, out_alloc);
}